// FlowNetC_61314953118059
// MI455X (gfx1250) — compile-verified
//
#include <hip/hip_runtime.h>
#include <stdint.h>

typedef unsigned int u32;
typedef u32 v4u __attribute__((ext_vector_type(4)));
typedef int  v4i __attribute__((ext_vector_type(4)));
typedef int  v8i __attribute__((ext_vector_type(8)));

#define Himg 512
#define Wimg 512
#define Bimg 16
#define TILE 32
#define X1S  34   // x1 LDS tile: 32 + 2 (conv halo)
#define X2S  38   // x2 LDS tile: 32 + 2*(corr halo 2 + conv halo 1)
#define F2S  36   // f2 LDS tile: 32 + 2*corr halo

#ifndef __has_builtin
#define __has_builtin(x) 0
#endif
#define HAVE_TDM __has_builtin(__builtin_amdgcn_tensor_load_to_lds)

__device__ __forceinline__ u32 lds_byte_off(const void* p) {
  // generic pointers into LDS carry the LDS byte offset in the low 32 bits
  return (u32)(uintptr_t)p;
}

#if HAVE_TDM
// Issue one TDM tile load: one channel plane, clamped to the image, written
// into a fixed-stride LDS buffer via descriptor-iteration (one row per iter).
__device__ __forceinline__ void tdm_load_tile(const float* gchan,
                                              u32 lds_base_bytes,
                                              int xs, int ys,    // unclamped tile origin
                                              int tsz,           // tile extent (== LDS stride)
                                              int lds_stride) {
  int xc = xs < 0 ? 0 : xs;
  int yc = ys < 0 ? 0 : ys;
  int xe = xs + tsz; if (xe > Wimg) xe = Wimg;
  int ye = ys + tsz; if (ye > Himg) ye = Himg;
  int tw = xe - xc;
  int th = ye - yc;
  u32 lds = lds_base_bytes + (u32)(((yc - ys) * lds_stride + (xc - xs)) * 4);
  uint64_t ga = (uint64_t)(uintptr_t)(gchan + (size_t)yc * Wimg + xc);

  v4u g0;
  g0[0] = 1u;                                        // count=1 (valid user descriptor)
  g0[1] = lds;                                       // lds_addr (bytes)
  g0[2] = (u32)(ga & 0xffffffffu);                   // global_addr[31:0]
  g0[3] = (u32)((ga >> 32) & 0x01ffffffu) | (2u << 30); // global_addr[56:32] | type=2

  v8i g1;
  g1[0] = (int)((2u << 16) | (1u << 19));            // data_size=4B, iterate_enable=1
  g1[1] = (int)(((u32)tw & 0xffffu) << 16);          // tensor_dim0[15:0]
  g1[2] = (int)(((u32)th & 0xffffu) << 16);          // dim0[31:16]=0 | tensor_dim1[15:0]
  g1[3] = (int)(((u32)tw) << 16);                    // dim1[31:16]=0 | tile_dim0=tw
  g1[4] = 1;                                         // tile_dim1=1 row/iter, tile_dim2=0
  g1[5] = Wimg;                                      // tensor_dim0_stride[31:0]
  g1[6] = (int)(((u32)Wimg & 0xffffu) << 16);        // dim0_stride[47:32]=0 | dim1_stride[15:0]
  g1[7] = 0;                                         // dim1_stride[47:16]

  v4i g2;
  g2[0] = 1;                                         // tensor_dim2 (unused)
  g2[1] = lds_stride;                                // lds_addr_increment (elements)
  g2[2] = Wimg;                                      // global_addr_increment[31:0] (elements)
  g2[3] = (int)(((u32)(th - 1)) << 16);              // inc[47:32]=0 | iterate_count=th-1
  v4i g3 = {0, 0, 0, 0};

#if __clang_major__ >= 23
  v8i g4 = {0, 0, 0, 0, 0, 0, 0, 0};
  __builtin_amdgcn_tensor_load_to_lds(g0, g1, g2, g3, g4, 0);
#else
  __builtin_amdgcn_tensor_load_to_lds(g0, g1, g2, g3, 0);
#endif
}
#else
// Fallback: cooperative vector-unit loads (used only if TDM builtin missing).
__device__ __forceinline__ void coop_load_tile(const float* gchan, float* lds,
                                               int xs, int ys, int tsz, int tid) {
  for (int i = tid; i < tsz * tsz; i += 256) {
    int r = i / tsz, c = i - r * tsz;
    int gy = ys + r, gx = xs + c;
    if (gy >= 0 && gy < Himg && gx >= 0 && gx < Wimg)
      lds[r * tsz + c] = gchan[(size_t)gy * Wimg + gx];
  }
}
#endif

__global__ __launch_bounds__(256) void flownetc_fused(
    const float* __restrict__ x1, const float* __restrict__ x2,
    const float* __restrict__ cw, float* __restrict__ out) {
  __shared__ __align__(16) float sW[96];             // w[k][o], k = c*9+ky*3+kx
  __shared__ __align__(16) float sX1[3 * X1S * X1S]; // 3468 (mult of 4)
  __shared__ __align__(16) float sX2[3 * X2S * X2S]; // 4332 (mult of 4)
  __shared__ __align__(16) float sF2[3 * F2S * F2S]; // 3888 (mult of 4)

  const int tid = threadIdx.x;
  const int x0 = blockIdx.x * TILE;
  const int y0 = blockIdx.y * TILE;
  const int b  = blockIdx.z;
  const size_t plane = (size_t)Himg * Wimg;

  // Block-uniform, branchless: does any tile halo touch the image border?
  const bool edge = ((x0 == 0) | (y0 == 0) |
                     (x0 == Wimg - TILE) | (y0 == Himg - TILE)) != 0;

  // Zero-fill input tiles only on edge blocks (interior tiles are fully
  // overwritten by the DMA). Implements conv zero-padding.
  if (edge) {
    const float4 z4 = make_float4(0.f, 0.f, 0.f, 0.f);
    float4* z1 = (float4*)sX1;
    float4* z2 = (float4*)sX2;
    for (int i = tid; i < (3 * X1S * X1S) / 4; i += 256) z1[i] = z4;
    for (int i = tid; i < (3 * X2S * X2S) / 4; i += 256) z2[i] = z4;
  }
  if (tid < 81) {
    int o = tid / 27, k = tid - o * 27;
    sW[k * 3 + o] = cw[tid];
  }
  __syncthreads();

#if HAVE_TDM
  // Scalarized guard: provably uniform, so exactly one wave issues the DMAs
  // (TDM ignores EXEC, so an exec-masked fall-through would duplicate them).
  if (__builtin_amdgcn_readfirstlane(tid) == 0) {
    #pragma unroll
    for (int c = 0; c < 3; ++c) {
      tdm_load_tile(x1 + ((size_t)b * 3 + c) * plane,
                    lds_byte_off(&sX1[c * X1S * X1S]), x0 - 1, y0 - 1, X1S, X1S);
      tdm_load_tile(x2 + ((size_t)b * 3 + c) * plane,
                    lds_byte_off(&sX2[c * X2S * X2S]), x0 - 3, y0 - 3, X2S, X2S);
    }
    __builtin_amdgcn_s_wait_tensorcnt(0);
  }
#else
  for (int c = 0; c < 3; ++c) {
    coop_load_tile(x1 + ((size_t)b * 3 + c) * plane, &sX1[c * X1S * X1S],
                   x0 - 1, y0 - 1, X1S, tid);
    coop_load_tile(x2 + ((size_t)b * 3 + c) * plane, &sX2[c * X2S * X2S],
                   x0 - 3, y0 - 3, X2S, tid);
  }
#endif
  __syncthreads();

  // Cache the 81 weights in VGPRs.
  float w[81];
  #pragma unroll
  for (int i = 0; i < 81; ++i) w[i] = sW[i];

  // ---- f2 = conv(x2) over the 36x36 halo region, 4-px strips -------------
  // 324 strips of 4 consecutive x; shared window loads (6 per (c,ky) per strip).
  for (int s = tid; s < (F2S / 4) * F2S; s += 256) {
    int yy = s / 9;
    int xb = (s - yy * 9) * 4;
    float a[3][4];
    #pragma unroll
    for (int o = 0; o < 3; ++o)
      #pragma unroll
      for (int j = 0; j < 4; ++j) a[o][j] = 0.f;

    #pragma unroll
    for (int c = 0; c < 3; ++c)
      #pragma unroll
      for (int ky = 0; ky < 3; ++ky) {
        float v[6];
        #pragma unroll
        for (int m = 0; m < 6; ++m)
          v[m] = sX2[(c * X2S + yy + ky) * X2S + xb + m];
        #pragma unroll
        for (int kx = 0; kx < 3; ++kx) {
          int k = (c * 3 + ky) * 3 + kx;
          #pragma unroll
          for (int o = 0; o < 3; ++o)
            #pragma unroll
            for (int j = 0; j < 4; ++j)
              a[o][j] = fmaf(v[kx + j], w[k * 3 + o], a[o][j]);
        }
      }

    if (edge) {
      // Zero outside the image (reference pads the conv OUTPUT before shifting).
      int gy = y0 - 2 + yy;
      bool rowin = (gy >= 0) & (gy < Himg);
      #pragma unroll
      for (int o = 0; o < 3; ++o) {
        float4 st;
        #pragma unroll
        for (int j = 0; j < 4; ++j) {
          int gx = x0 - 2 + xb + j;
          bool in = rowin & (gx >= 0) & (gx < Wimg);
          (&st.x)[j] = in ? a[o][j] : 0.f;
        }
        *(float4*)&sF2[o * F2S * F2S + yy * F2S + xb] = st;
      }
    } else {
      #pragma unroll
      for (int o = 0; o < 3; ++o) {
        float4 st;
        #pragma unroll
        for (int j = 0; j < 4; ++j) (&st.x)[j] = a[o][j];
        *(float4*)&sF2[o * F2S * F2S + yy * F2S + xb] = st;
      }
    }
  }
  __syncthreads();

  // ---- f1 = conv(x1): exactly one 4-px strip per thread, kept in regs ----
  const int y  = tid >> 3;         // 0..31
  const int xb = (tid & 7) * 4;    // 0,4,...,28
  float f1[3][4];
  #pragma unroll
  for (int o = 0; o < 3; ++o)
    #pragma unroll
    for (int j = 0; j < 4; ++j) f1[o][j] = 0.f;

  #pragma unroll
  for (int c = 0; c < 3; ++c)
    #pragma unroll
    for (int ky = 0; ky < 3; ++ky) {
      float v[6];
      #pragma unroll
      for (int m = 0; m < 6; ++m)
        v[m] = sX1[(c * X1S + y + ky) * X1S + xb + m];
      #pragma unroll
      for (int kx = 0; kx < 3; ++kx) {
        int k = (c * 3 + ky) * 3 + kx;
        #pragma unroll
        for (int o = 0; o < 3; ++o)
          #pragma unroll
          for (int j = 0; j < 4; ++j)
            f1[o][j] = fmaf(v[kx + j], w[k * 3 + o], f1[o][j]);
      }
    }

  // Fold the channel-mean (1/3) into f1 once: correlation becomes pure FMAs.
  #pragma unroll
  for (int o = 0; o < 3; ++o)
    #pragma unroll
    for (int j = 0; j < 4; ++j) f1[o][j] *= (1.0f / 3.0f);

  // ---- correlation: 9 displacements, mean over 3 channels ----------------
  #pragma unroll
  for (int dyi = 0; dyi < 3; ++dyi) {
    int ry = y + dyi * 2;          // sF2 row: (y+dy)+2 with dy in {-2,0,2}
    float g[3][8];                 // x window xb..xb+7 covers dx 0,2,4 + j 0..3
    #pragma unroll
    for (int o = 0; o < 3; ++o)
      #pragma unroll
      for (int m = 0; m < 8; ++m)
        g[o][m] = sF2[o * F2S * F2S + ry * F2S + xb + m];

    #pragma unroll
    for (int dxi = 0; dxi < 3; ++dxi) {
      int d = dyi * 3 + dxi;
      int ox = dxi * 2;
      float4 r;
      #pragma unroll
      for (int j = 0; j < 4; ++j)
        (&r.x)[j] = fmaf(f1[0][j], g[0][ox + j],
                    fmaf(f1[1][j], g[1][ox + j],
                         f1[2][j] * g[2][ox + j]));
      *(float4*)&out[(((size_t)b * 9 + d) * Himg + (y0 + y)) * Wimg + x0 + xb] = r;
    }
  }
}

extern "C" void kernel_launch(void* const* d_in, const int* in_sizes, int n_in,
                              void* d_out, int out_size, void* d_ws, size_t ws_size,
                              hipStream_t stream) {
  const float* x1 = (const float*)d_in[0];
  const float* x2 = (const float*)d_in[1];
  const float* cw = (const float*)d_in[2];
  float* out = (float*)d_out;
  dim3 grid(Wimg / TILE, Himg / TILE, Bimg);
  flownetc_fused<<<grid, dim3(256), 0, stream>>>(x1, x2, cw, out);
}